// SoftHd_71648644432487
// MI455X (gfx1250) — compile-verified
//
#include <hip/hip_runtime.h>
#include <math.h>

// ---------------------------------------------------------------------------
// Soft Hausdorff distance, fused one-pass kernel for MI455X (gfx1250).
//   v1: (64,128,256) f32, v2: (64,128,256) f32, sz1/sz2: (64,) i32 -> out (64,64) f32
// One workgroup (256 threads = 8 wave32) per (i,j) graph pair.
// GEMM core: V_WMMA_F32_16X16X4_F32 on double-buffered LDS K-chunks fed by
// GLOBAL_LOAD_ASYNC_TO_LDS_B128 (ASYNCcnt-pipelined).
// ---------------------------------------------------------------------------

typedef __attribute__((ext_vector_type(2))) float v2f;
typedef __attribute__((ext_vector_type(8))) float v8f;

#define LNODES 128          // L1 == L2
#define FDIM   256          // feature dim
#define KC     32           // K-chunk staged in LDS
#define NCHUNK (FDIM / KC)  // 8
#define PITCH  (KC + 4)     // 36 floats: 144B rows -> 16B aligned, conflict-free b64 reads
#define EPSF   1e-12f

// async global->LDS copy, 16B per lane (GVS mode: sgpr base + 32-bit vgpr offset)
__device__ __forceinline__ void async_ld16(const float* __restrict__ base,
                                           unsigned goff, unsigned lds_addr) {
    asm volatile("global_load_async_to_lds_b128 %0, %1, %2"
                 :
                 : "v"(lds_addr), "v"(goff), "s"(base)
                 : "memory");
}
__device__ __forceinline__ void wait_async_le8() {
    asm volatile("s_wait_asynccnt 0x8" ::: "memory");
}
__device__ __forceinline__ void wait_async_le0() {
    asm volatile("s_wait_asynccnt 0x0" ::: "memory");
}

__global__ __launch_bounds__(256) void softhd_kernel(
    const float* __restrict__ v1, const float* __restrict__ v2,
    const int* __restrict__ sz1, const int* __restrict__ sz2,
    float* __restrict__ out, int N2)
{
    __shared__ float sA[2][LNODES][PITCH];   // double-buffered v1 rows x K-chunk
    __shared__ float sB[2][LNODES][PITCH];   // double-buffered v2 rows x K-chunk
    __shared__ float x2s[LNODES];
    __shared__ float y2s[LNODES];
    __shared__ int   colminb[LNODES];        // min over b per v2-node a (float bits)
    __shared__ int   rowminb[LNODES];        // min over a per v1-node b (float bits)
    __shared__ float wsum[8];

    const int i    = blockIdx.y;
    const int j    = blockIdx.x;
    const int tid  = threadIdx.x;
    const int w    = tid >> 5;               // wave id 0..7
    const int lane = tid & 31;
    const int ln   = lane & 15;              // N / M index within half-wave
    const int hl   = lane >> 4;              // half-wave select
    const int szi  = sz1[i];
    const int szj  = sz2[j];

    if (tid < LNODES) {
        colminb[tid] = 0x7f800000;           // +inf bits
        rowminb[tid] = 0x7f800000;
    }

    const float* __restrict__ A = v1 + (size_t)i * LNODES * FDIM;
    const float* __restrict__ B = v2 + (size_t)j * LNODES * FDIM;

    // accumulators: 8 tiles of 16x16 along a; this wave owns rows 16w..16w+15
    v8f acc[8];
    #pragma unroll
    for (int t = 0; t < 8; ++t)
        #pragma unroll
        for (int q = 0; q < 8; ++q) acc[t][q] = 0.0f;

    // async copy mapping: 8 float4 per row, 32 rows per pass, 4 passes per array
    const int lrow = tid >> 3;               // 0..31
    const int lcol = (tid & 7) * 4;          // 0,4,...,28

    auto issue_chunk = [&](int c, int buf) {
        #pragma unroll
        for (int p = 0; p < 4; ++p) {
            const int row = lrow + 32 * p;
            const unsigned goff =
                (unsigned)(((unsigned)row * FDIM + (unsigned)(c * KC) + (unsigned)lcol)
                           * sizeof(float));
            async_ld16(A, goff, (unsigned)(size_t)&sA[buf][row][lcol]);
            async_ld16(B, goff, (unsigned)(size_t)&sB[buf][row][lcol]);
        }
    };

    // per-row squared-norm partials (one owner thread per row; wave-uniform split)
    const int nrow = tid & (LNODES - 1);     // row this thread owns for norms
    float nacc = 0.0f;

    issue_chunk(0, 0);                       // pipeline prologue

    for (int c = 0; c < NCHUNK; ++c) {
        const int cur = c & 1;
        if (c + 1 < NCHUNK) {
            issue_chunk(c + 1, cur ^ 1);     // prefetch next chunk into other buffer
            wait_async_le8();                // previous 8 loads (chunk c) have landed
        } else {
            wait_async_le0();
        }
        __syncthreads();                     // chunk c visible to all waves

        const float (*cA)[PITCH] = sA[cur];
        const float (*cB)[PITCH] = sB[cur];

        // f32 WMMA GEMM core
        #pragma unroll 2
        for (int kk = 0; kk < KC / 4; ++kk) {
            const int k = 4 * kk + 2 * hl;   // per-lane K offset (A/B operand layout)
            const v2f a = *(const v2f*)&cA[16 * w + ln][k];
            #pragma unroll
            for (int t = 0; t < 8; ++t) {
                const v2f b = *(const v2f*)&cB[16 * t + ln][k];
                acc[t] = __builtin_amdgcn_wmma_f32_16x16x4_f32(
                    false, a, false, b, (short)0, acc[t], false, false);
            }
        }

        // squared-norm partials from LDS (waves 0-3: v1 rows, waves 4-7: v2 rows)
        if (tid < LNODES) {
            #pragma unroll
            for (int k0 = 0; k0 < KC; ++k0) {
                const float x = cA[nrow][(nrow + k0) & (KC - 1)]; // staggered: no bank camp
                nacc += x * x;
            }
        } else {
            #pragma unroll
            for (int k0 = 0; k0 < KC; ++k0) {
                const float x = cB[nrow][(nrow + k0) & (KC - 1)];
                nacc += x * x;
            }
        }
        __syncthreads();                     // compute done before buffer reuse
    }

    if (tid < LNODES) x2s[nrow] = nacc; else y2s[nrow] = nacc;
    __syncthreads();

    // epilogue: d = sqrt(max(x2 + y2 - 2*xy, eps)); masked row/col mins
    float xr[8], yc[8];
    #pragma unroll
    for (int r = 0; r < 8; ++r) xr[r] = x2s[16 * w + 8 * hl + r];
    #pragma unroll
    for (int t = 0; t < 8; ++t) yc[t] = y2s[16 * t + ln];

    const float INF = __int_as_float(0x7f800000);
    float rowcand[8];
    #pragma unroll
    for (int r = 0; r < 8; ++r) rowcand[r] = INF;

    #pragma unroll
    for (int t = 0; t < 8; ++t) {
        const int a = 16 * t + ln;
        const bool va = (a < szj);
        float colc = INF;
        #pragma unroll
        for (int r = 0; r < 8; ++r) {
            const float d2 = xr[r] + yc[t] - 2.0f * acc[t][r];
            const float d  = sqrtf(fmaxf(d2, EPSF));
            const int  b   = 16 * w + 8 * hl + r;
            if (b < szi) colc = fminf(colc, d);
            if (va) rowcand[r] = fminf(rowcand[r], d);
        }
        if (colc < INF) atomicMin(&colminb[a], __float_as_int(colc)); // d>=0: int order == float order
    }
    #pragma unroll
    for (int r = 0; r < 8; ++r) {
        if (rowcand[r] < INF)
            atomicMin(&rowminb[16 * w + 8 * hl + r], __float_as_int(rowcand[r]));
    }
    __syncthreads();

    // final block sum: valid colmins + valid rowmins
    float v = 0.0f;
    if (tid < LNODES) {
        if (tid < szj) v = __int_as_float(colminb[tid]);
    } else {
        const int b = tid - LNODES;
        if (b < szi) v = __int_as_float(rowminb[b]);
    }
    #pragma unroll
    for (int off = 16; off > 0; off >>= 1) v += __shfl_down(v, off, 32);
    if (lane == 0) wsum[w] = v;
    __syncthreads();
    if (tid == 0) {
        float s = 0.f;
        #pragma unroll
        for (int q = 0; q < 8; ++q) s += wsum[q];
        out[(size_t)i * N2 + j] = s;
    }
}

extern "C" void kernel_launch(void* const* d_in, const int* in_sizes, int n_in,
                              void* d_out, int out_size, void* d_ws, size_t ws_size,
                              hipStream_t stream) {
    // setup_inputs order: v1, am1, sz1, v2, am2, sz2
    const float* v1  = (const float*)d_in[0];
    const int*   sz1 = (const int*)  d_in[2];
    const float* v2  = (const float*)d_in[3];
    const int*   sz2 = (const int*)  d_in[5];
    float* out = (float*)d_out;

    const int N1 = in_sizes[2];
    const int N2 = in_sizes[5];

    dim3 grid(N2, N1, 1);
    dim3 block(256, 1, 1);
    softhd_kernel<<<grid, block, 0, stream>>>(v1, v2, sz1, sz2, out, N2);
}